// MFGN_69836168233308
// MI455X (gfx1250) — compile-verified
//
#include <hip/hip_runtime.h>
#include <math.h>

// ---------------- problem constants ----------------
constexpr int O_  = 2048;
constexpr int NI_ = 16;
constexpr int F_  = 4;
constexpr int D_  = 128;
constexpr int D2_ = 64;
constexpr int NN_ = 8;
constexpr int LDR = 132;   // padded LDS row stride (floats) -> conflict-free column walks

typedef float v2f __attribute__((ext_vector_type(2)));
typedef float v8f __attribute__((ext_vector_type(8)));

// ---------------- CDNA5 fp32 WMMA helpers ----------------
__device__ __forceinline__ v8f wmma_f32(v2f a, v2f b, v8f c) {
  // D = A(16x4,f32) * B(4x16,f32) + C(16x16,f32)
  return __builtin_amdgcn_wmma_f32_16x16x4_f32(false, a, false, b, (short)0, c,
                                               false, false);
}

// B operand staged in LDS in "k-quad" layout WQ[k/4][n][4]:
//   WQ[((k>>2)*N + n)*4 + (k&3)] = W[k*N + n]
// so each lane's (ka, ka+1) pair is one aligned 8-byte LDS load, and the
// dword bank index is (4*idx + 2*half + const) -> conflict-free for 32 lanes.
template <int K, int N>
__device__ __forceinline__ v8f gemm_tile_q(const float* __restrict__ X, int ldx,
                                           const float* __restrict__ WQ,
                                           int ncol0, int lane) {
  const int half = lane >> 4;   // 0: K{0,1}, 1: K{2,3} within each k-step
  const int idx  = lane & 15;   // m for A, n for B
  const float* xp = X + idx * ldx + 2 * half;
  const float* wp = WQ + (ncol0 + idx) * 4 + 2 * half;
  v8f c = {};
#pragma unroll
  for (int k = 0; k < K; k += 4) {
    v2f a = *(const v2f*)(xp + k);
    v2f b = *(const v2f*)(wp + (k >> 2) * (N * 4));
    c = wmma_f32(a, b, c);
  }
  return c;
}

// B operand from global, row-major [K][ldw], UNIFORM base (kernel argument) so
// the backend emits saddr-form loads with immediate k-step offsets.
template <int K>
__device__ __forceinline__ v8f gemm_tile_g(const float* __restrict__ X, int ldx,
                                           const float* __restrict__ W, int ldw,
                                           int ncol0, int lane) {
  const int half = lane >> 4, idx = lane & 15;
  const float* xp = X + idx * ldx + 2 * half;
  const float* wp = W + (2 * half) * ldw + ncol0 + idx;
  v8f c = {};
#pragma unroll
  for (int k = 0; k < K; k += 4) {
    v2f a = *(const v2f*)(xp + k);
    v2f b = { wp[k * ldw], wp[(k + 1) * ldw] };
    c = wmma_f32(a, b, c);
  }
  return c;
}

__device__ __forceinline__ float lrelu(float v) { return v > 0.f ? v : 0.01f * v; }

// store a C tile with bias add + leaky-relu into LDS (row-major, stride ldd)
__device__ __forceinline__ void store_tile_lrelu(float* __restrict__ Dst, int ldd,
                                                 int ncol0,
                                                 const float* __restrict__ bias,
                                                 v8f c, int lane) {
  const int half = lane >> 4, idx = lane & 15;
  const float bv = bias[ncol0 + idx];
#pragma unroll
  for (int r = 0; r < 8; ++r) {
    const int m = r + 8 * half;
    Dst[m * ldd + ncol0 + idx] = lrelu(c[r] + bv);
  }
}

// contiguous global -> LDS copy, float4 granularity (ds_store_b128)
__device__ __forceinline__ void copy_to_lds(float* __restrict__ dst,
                                            const float* __restrict__ src,
                                            int nfloats, int tid) {
  const float4* s = (const float4*)src;
  float4* d = (float4*)dst;
  for (int t = tid; t < (nfloats >> 2); t += 256) d[t] = s[t];
}

// =====================================================================
// Kernel 0: repack weights into k-quad layout (once per launch, tiny)
// =====================================================================
__global__ void k_pack(const float* __restrict__ src, float* __restrict__ dst,
                       int K, int N, int B) {
  const int total = B * K * N;
  for (int t = blockIdx.x * blockDim.x + threadIdx.x; t < total;
       t += gridDim.x * blockDim.x) {
    const int b = t / (K * N);
    const int r = t - b * K * N;
    const int k = r / N, n = r - k * N;
    dst[b * K * N + ((k >> 2) * N + n) * 4 + (k & 3)] = src[t];
  }
}

// =====================================================================
// Kernel 1: creat_factors  fac[o,s,f,:] = lrelu(lrelu(x W1_f + b1_f) W2_f + b2_f)
// one block per outfit, 256 threads = 8 waves; weights staged 2 factors/phase
// =====================================================================
__global__ void k_creat_factors(const float* __restrict__ items_feature,
                                const float* __restrict__ wq_cf1,  // packed [F][128][64]
                                const float* __restrict__ cf_b1,
                                const float* __restrict__ wq_cf2,  // packed [F][64][128]
                                const float* __restrict__ cf_b2,
                                float* __restrict__ fac) {
  const int o = blockIdx.x, tid = threadIdx.x;
  const int wave = tid >> 5, lane = tid & 31;
  __shared__ float WQ[2 * D_ * D2_];     // 64KB: two factors of one layer
  __shared__ float X[NI_][LDR];          // 16 x 128 item features
  __shared__ float H[F_][NI_][D2_ + 4];  // per-factor hidden 16 x 64

  for (int t = tid; t < NI_ * D_; t += 256) {
    const int r = t >> 7, d = t & 127;
    X[r][d] = items_feature[(size_t)(o * NI_ + r) * D_ + d];
  }

  // ---- layer 1 (two phases of 2 factors) ----
  for (int p = 0; p < 2; ++p) {
    __syncthreads();
    copy_to_lds(WQ, wq_cf1 + p * 2 * D_ * D2_, 2 * D_ * D2_, tid);
    __syncthreads();
    const int fl = wave >> 2, nt = wave & 3, f = p * 2 + fl;  // 8 jobs = 8 waves
    v8f c = gemm_tile_q<D_, D2_>(&X[0][0], LDR, WQ + fl * D_ * D2_, nt * 16, lane);
    store_tile_lrelu(&H[f][0][0], D2_ + 4, nt * 16, cf_b1 + f * D2_, c, lane);
  }

  // ---- layer 2 (two phases of 2 factors) ----
  for (int p = 0; p < 2; ++p) {
    __syncthreads();
    copy_to_lds(WQ, wq_cf2 + p * 2 * D2_ * D_, 2 * D2_ * D_, tid);
    __syncthreads();
    for (int job = wave; job < 16; job += 8) {  // 2 factors x 8 ntiles
      const int fl = job >> 3, nt = job & 7, f = p * 2 + fl;
      v8f c = gemm_tile_q<D2_, D_>(&H[f][0][0], D2_ + 4, WQ + fl * D2_ * D_,
                                   nt * 16, lane);
      const int half = lane >> 4, idx = lane & 15;
      const float bv = cf_b2[f * D_ + nt * 16 + idx];
#pragma unroll
      for (int r = 0; r < 8; ++r) {
        const int s = r + 8 * half;  // slot
        fac[(((size_t)o * NI_ + s) * F_ + f) * D_ + nt * 16 + idx] = lrelu(c[r] + bv);
      }
    }
  }
}

// =====================================================================
// Kernel 2: inter_factors (sequential per outfit, fac in LDS) + com partial
// W1 staged in LDS (packed); W2 streamed from global (uniform base, L1-hot)
// =====================================================================
__global__ void k_inter_factors(const int* __restrict__ outfit_items,
                                const int* __restrict__ items_neighbor,
                                const float* __restrict__ wq_W1,
                                const float* __restrict__ b1,
                                const float* __restrict__ W2,
                                const float* __restrict__ b2,
                                float* __restrict__ fac,
                                float* __restrict__ com_partial) {
  const int o = blockIdx.x, tid = threadIdx.x;
  const int wave = tid >> 5, lane = tid & 31;
  __shared__ float WQ[D_ * D_];       // 64KB packed W1
  __shared__ float Fa[NI_][F_][LDR];  // outfit factors (updated in place)
  __shared__ float Xb[16][LDR];       // chunk input / reused as message buffer
  __shared__ float Hb[16][LDR];       // chunk hidden
  __shared__ float Tt[F_][D_];        // snapshot of fac[ic]
  __shared__ float Acc[F_][D_];       // accumulated messages for fac[ic]
  __shared__ int oi[NI_];
  __shared__ int nbr[NI_ * NN_];
  __shared__ float red[256];

  copy_to_lds(WQ, wq_W1, D_ * D_, tid);
  for (int t = tid; t < NI_ * F_ * D_; t += 256) {
    const int s = t >> 9, f = (t >> 7) & 3, d = t & 127;
    Fa[s][f][d] = fac[(((size_t)o * NI_ + s) * F_ + f) * D_ + d];
  }
  if (tid < NI_) oi[tid] = outfit_items[o * NI_ + tid];
  for (int t = tid; t < NI_ * NN_; t += 256) nbr[t] = items_neighbor[o * NI_ * NN_ + t];
  __syncthreads();

  for (int s = 0; s < NI_; ++s) {
    const int i = oi[s];          // uniform across block
    if (i == -1) continue;        // uniform branch
    const int ic = i;
    for (int t = tid; t < F_ * D_; t += 256) {
      const int f = t >> 7, d = t & 127;
      Tt[f][d] = Fa[ic][f][d];
      Acc[f][d] = 0.f;
    }
    __syncthreads();

    // 24 candidates (16 outfit slots + 8 neighbors of ic) in 6 chunks of 4;
    // rows r = cand_local*4 + factor  -> one 16-row M tile per chunk
    for (int ch = 0; ch < 6; ++ch) {
      for (int t = tid; t < 16 * D_; t += 256) {
        const int r = t >> 7, d = t & 127;
        const int cl = r >> 2, f = r & 3;
        const int cidx = ch * 4 + cl;
        const int cand = (cidx < NI_) ? oi[cidx] : nbr[ic * NN_ + (cidx - NI_)];
        const int g = cand < 0 ? 0 : cand;
        Xb[r][d] = Tt[f][d] * Fa[g][f][d];
      }
      __syncthreads();
      {  // layer 1: 8 N-tiles, one per wave; B from LDS quad layout
        v8f c = gemm_tile_q<D_, D_>(&Xb[0][0], LDR, WQ, wave * 16, lane);
        store_tile_lrelu(&Hb[0][0], LDR, wave * 16, b1, c, lane);
      }
      __syncthreads();
      {  // layer 2 -> messages back into Xb; B from global (uniform base)
        v8f c = gemm_tile_g<D_>(&Hb[0][0], LDR, W2, D_, wave * 16, lane);
        store_tile_lrelu(&Xb[0][0], LDR, wave * 16, b2, c, lane);
      }
      __syncthreads();
      // masked accumulate over the 4 candidates of this chunk
      for (int t = tid; t < F_ * D_; t += 256) {
        const int f = t >> 7, d = t & 127;
        float a = 0.f;
#pragma unroll
        for (int cl = 0; cl < 4; ++cl) {
          const int cidx = ch * 4 + cl;
          const int cand = (cidx < NI_) ? oi[cidx] : nbr[ic * NN_ + (cidx - NI_)];
          const bool ok = (cand != -1) && (cand != i);
          a += ok ? Xb[cl * 4 + f][d] : 0.f;
        }
        Acc[f][d] += a;
      }
      __syncthreads();
    }
    for (int t = tid; t < F_ * D_; t += 256) {
      const int f = t >> 7, d = t & 127;
      Fa[ic][f][d] += Acc[f][d];
    }
    __syncthreads();
  }

  // write updated fac back
  for (int t = tid; t < NI_ * F_ * D_; t += 256) {
    const int s = t >> 9, f = (t >> 7) & 3, d = t & 127;
    fac[(((size_t)o * NI_ + s) * F_ + f) * D_ + d] = Fa[s][f][d];
  }

  // com_loss partial: sum over valid slots of ||Fa[s] Fa[s]^T - I||_F^2
  float part = 0.f;
  {
    const int s = tid >> 4, f = (tid >> 2) & 3, g = tid & 3;  // 256 = 16*4*4 combos
    if (oi[s] != -1) {
      float dot = 0.f;
      for (int d = 0; d < D_; ++d) dot += Fa[s][f][d] * Fa[s][g][d];
      const float G = dot - (f == g ? 1.f : 0.f);
      part = G * G;
    }
  }
  red[tid] = part;
  __syncthreads();
  for (int st = 128; st > 0; st >>= 1) {
    if (tid < st) red[tid] += red[tid + st];
    __syncthreads();
  }
  if (tid == 0) com_partial[o] = red[0];
}

// =====================================================================
// Kernel 3: infer_items  feat = items_feature + scatter(sum_f f2i(fac))
// =====================================================================
__global__ void k_infer_items(const int* __restrict__ outfit_items,
                              const float* __restrict__ items_feature,
                              const float* __restrict__ fac,
                              const float* __restrict__ wq_W1,
                              const float* __restrict__ b1,
                              const float* __restrict__ W2,
                              const float* __restrict__ b2,
                              float* __restrict__ feat) {
  const int o = blockIdx.x, tid = threadIdx.x;
  const int wave = tid >> 5, lane = tid & 31;
  __shared__ float WQ[D_ * D_];  // 64KB packed W1
  __shared__ float Xb[16][LDR], Hb[16][LDR];
  __shared__ float Gs[NI_][D_];  // per-slot message sum over factors
  __shared__ float Ft[NI_][D_];  // output features
  __shared__ int oi[NI_];

  copy_to_lds(WQ, wq_W1, D_ * D_, tid);
  if (tid < NI_) oi[tid] = outfit_items[o * NI_ + tid];
  for (int t = tid; t < NI_ * D_; t += 256) {
    const int r = t >> 7, d = t & 127;
    Gs[r][d] = 0.f;
    Ft[r][d] = items_feature[(size_t)(o * NI_ + r) * D_ + d];
  }
  __syncthreads();

  // fac viewed as [64,128]; 4 M-chunks of 16 rows (4 slots x 4 factors)
  for (int mc = 0; mc < 4; ++mc) {
    for (int t = tid; t < 16 * D_; t += 256) {
      const int r = t >> 7, d = t & 127;
      Xb[r][d] = fac[((size_t)o * 64 + mc * 16 + r) * D_ + d];
    }
    __syncthreads();
    v8f c = gemm_tile_q<D_, D_>(&Xb[0][0], LDR, WQ, wave * 16, lane);
    store_tile_lrelu(&Hb[0][0], LDR, wave * 16, b1, c, lane);
    __syncthreads();
    c = gemm_tile_g<D_>(&Hb[0][0], LDR, W2, D_, wave * 16, lane);
    {  // lrelu + sum over the 4 factor-rows of each slot
      const int half = lane >> 4, idx = lane & 15;
      const float bv = b2[wave * 16 + idx];
      float a0 = 0.f, a1 = 0.f;
#pragma unroll
      for (int r = 0; r < 4; ++r) a0 += lrelu(c[r] + bv);
#pragma unroll
      for (int r = 4; r < 8; ++r) a1 += lrelu(c[r] + bv);
      const int s0 = mc * 4 + (half ? 2 : 0);  // rows m=0..3 -> slot0, 4..7 -> slot1
      Gs[s0][wave * 16 + idx] += a0;
      Gs[s0 + 1][wave * 16 + idx] += a1;
    }
    __syncthreads();
  }

  // scatter: Ft[r] += (#valid slots with item==r) * Gs[r]
  for (int t = tid; t < NI_ * D_; t += 256) {
    const int r = t >> 7, d = t & 127;
    int cnt = 0;
#pragma unroll
    for (int s = 0; s < NI_; ++s) cnt += (oi[s] == r) ? 1 : 0;
    Ft[r][d] += (float)cnt * Gs[r][d];
  }
  __syncthreads();
  for (int t = tid; t < NI_ * D_; t += 256) {
    const int r = t >> 7, d = t & 127;
    feat[(size_t)(o * NI_ + r) * D_ + d] = Ft[r][d];
  }
}

// =====================================================================
// Kernel 4: inter_items (sequential per outfit) + outfit score
// =====================================================================
__global__ void k_inter_items(const int* __restrict__ outfit_items,
                              const float* __restrict__ wq_W1,
                              const float* __restrict__ b1,
                              const float* __restrict__ W2,
                              const float* __restrict__ b2,
                              const float* __restrict__ o2s_W,
                              const float* __restrict__ o2s_b,
                              const float* __restrict__ feat,
                              float* __restrict__ scores) {
  const int o = blockIdx.x, tid = threadIdx.x;
  const int wave = tid >> 5, lane = tid & 31;
  __shared__ float WQ[D_ * D_];  // 64KB packed W1
  __shared__ float Ft[NI_][D_];
  __shared__ float Xb[16][LDR], Hb[16][LDR], Mg[16][LDR];
  __shared__ int oi[NI_];
  __shared__ float red[256];

  copy_to_lds(WQ, wq_W1, D_ * D_, tid);
  if (tid < NI_) oi[tid] = outfit_items[o * NI_ + tid];
  for (int t = tid; t < NI_ * D_; t += 256) {
    const int r = t >> 7, d = t & 127;
    Ft[r][d] = feat[(size_t)(o * NI_ + r) * D_ + d];
  }
  __syncthreads();

  for (int s = 0; s < NI_; ++s) {
    const int i = oi[s];
    if (i == -1) continue;  // uniform
    // gather X = Ft[max(oi,0)]
    for (int t = tid; t < 16 * D_; t += 256) {
      const int r = t >> 7, d = t & 127;
      const int j = oi[r];
      Xb[r][d] = Ft[j < 0 ? 0 : j][d];
    }
    __syncthreads();
    v8f c = gemm_tile_q<D_, D_>(&Xb[0][0], LDR, WQ, wave * 16, lane);
    store_tile_lrelu(&Hb[0][0], LDR, wave * 16, b1, c, lane);
    __syncthreads();
    c = gemm_tile_g<D_>(&Hb[0][0], LDR, W2, D_, wave * 16, lane);
    store_tile_lrelu(&Mg[0][0], LDR, wave * 16, b2, c, lane);
    __syncthreads();
    if (tid < D_) {  // Ft[i] = Ft[i] + sum_j mask_j * Mg[j]
      float a = 0.f;
#pragma unroll
      for (int j = 0; j < NI_; ++j) {
        const bool ok = (oi[j] != -1) && (oi[j] != i);
        a += ok ? Mg[j][tid] : 0.f;
      }
      Ft[i][tid] += a;
    }
    __syncthreads();
  }

  // score: sigmoid( (sum_s w_s Ft[ic_s]) . o2s_W + o2s_b )
  float val = 0.f;
  if (tid < D_) {
    float acc = 0.f;
#pragma unroll
    for (int s = 0; s < NI_; ++s) {
      const int i = oi[s];
      if (i != -1) acc += Ft[i][tid];
    }
    val = acc * o2s_W[tid];
  }
  red[tid] = val;
  __syncthreads();
  for (int st = 128; st > 0; st >>= 1) {
    if (tid < st) red[tid] += red[tid + st];
    __syncthreads();
  }
  if (tid == 0) scores[o] = 1.f / (1.f + expf(-(red[0] + o2s_b[0])));
}

// =====================================================================
// Kernel 5: deterministic fixed-order reduction of com partials
// =====================================================================
__global__ void k_reduce_com(const float* __restrict__ part, float* __restrict__ out_com) {
  __shared__ float red[256];
  const int tid = threadIdx.x;
  float s = 0.f;
  for (int i = tid; i < O_; i += 256) s += part[i];
  red[tid] = s;
  __syncthreads();
  for (int st = 128; st > 0; st >>= 1) {
    if (tid < st) red[tid] += red[tid + st];
    __syncthreads();
  }
  if (tid == 0) out_com[0] = red[0] * (1.0f / (float)O_);
}

// =====================================================================
extern "C" void kernel_launch(void* const* d_in, const int* in_sizes, int n_in,
                              void* d_out, int out_size, void* d_ws, size_t ws_size,
                              hipStream_t stream) {
  (void)in_sizes; (void)n_in; (void)out_size; (void)ws_size;
  const int*   outfit_items   = (const int*)  d_in[0];
  const float* items_feature  = (const float*)d_in[1];
  const int*   items_neighbor = (const int*)  d_in[2];
  // d_in[3] = items_factors (unused: fully overwritten by creat_factors)
  const float* cf_W1  = (const float*)d_in[4];
  const float* cf_b1  = (const float*)d_in[5];
  const float* cf_W2  = (const float*)d_in[6];
  const float* cf_b2  = (const float*)d_in[7];
  const float* f2f_W1 = (const float*)d_in[8];
  const float* f2f_b1 = (const float*)d_in[9];
  const float* f2f_W2 = (const float*)d_in[10];
  const float* f2f_b2 = (const float*)d_in[11];
  const float* f2i_W1 = (const float*)d_in[12];
  const float* f2i_b1 = (const float*)d_in[13];
  const float* f2i_W2 = (const float*)d_in[14];
  const float* f2i_b2 = (const float*)d_in[15];
  const float* i2i_W1 = (const float*)d_in[16];
  const float* i2i_b1 = (const float*)d_in[17];
  const float* i2i_W2 = (const float*)d_in[18];
  const float* i2i_b2 = (const float*)d_in[19];
  const float* o2s_W  = (const float*)d_in[20];
  const float* o2s_b  = (const float*)d_in[21];

  float* out  = (float*)d_out;                          // [0..2047]=scores, [2048]=com
  float* fac  = (float*)d_ws;                           // O*NI*F*D floats (64MB)
  float* feat = fac + (size_t)O_ * NI_ * F_ * D_;       // O*NI*D (16MB)
  float* comp = feat + (size_t)O_ * NI_ * D_;           // O floats
  float* wq_cf1  = comp + O_;                           // packed weights
  float* wq_cf2  = wq_cf1 + F_ * D_ * D2_;
  float* wq_f2f1 = wq_cf2 + F_ * D2_ * D_;
  float* wq_f2i1 = wq_f2f1 + D_ * D_;
  float* wq_i2i1 = wq_f2i1 + D_ * D_;

  k_pack<<<64, 256, 0, stream>>>(cf_W1, wq_cf1, D_, D2_, F_);
  k_pack<<<64, 256, 0, stream>>>(cf_W2, wq_cf2, D2_, D_, F_);
  k_pack<<<64, 256, 0, stream>>>(f2f_W1, wq_f2f1, D_, D_, 1);
  k_pack<<<64, 256, 0, stream>>>(f2i_W1, wq_f2i1, D_, D_, 1);
  k_pack<<<64, 256, 0, stream>>>(i2i_W1, wq_i2i1, D_, D_, 1);

  k_creat_factors<<<O_, 256, 0, stream>>>(items_feature, wq_cf1, cf_b1, wq_cf2,
                                          cf_b2, fac);
  k_inter_factors<<<O_, 256, 0, stream>>>(outfit_items, items_neighbor,
                                          wq_f2f1, f2f_b1, f2f_W2, f2f_b2, fac, comp);
  k_infer_items<<<O_, 256, 0, stream>>>(outfit_items, items_feature, fac,
                                        wq_f2i1, f2i_b1, f2i_W2, f2i_b2, feat);
  k_inter_items<<<O_, 256, 0, stream>>>(outfit_items, wq_i2i1, i2i_b1, i2i_W2,
                                        i2i_b2, o2s_W, o2s_b, feat, out);
  k_reduce_com<<<1, 256, 0, stream>>>(comp, out + O_);
}